// SGC_79577154060346
// MI455X (gfx1250) — compile-verified
//
#include <hip/hip_runtime.h>
#include <math.h>

typedef __attribute__((ext_vector_type(2))) float v2f;
typedef __attribute__((ext_vector_type(8))) float v8f;

#define D 128  // D_IN == D_OUT == 128

// Native f32 global atomic add (lowers to GLOBAL_ATOMIC_ADD_F32, non-returning).
__device__ __forceinline__ void atom_add_f32(float* p, float v) {
  unsafeAtomicAdd(p, v);
}

// ---------- normalization ----------
__global__ __launch_bounds__(256) void k_init_deg(float* __restrict__ deg, int n) {
  int i = blockIdx.x * blockDim.x + threadIdx.x;
  if (i < n) deg[i] = 1.0f;  // self-loop fill value
}

__global__ __launch_bounds__(256) void k_accum_deg(const int* __restrict__ col,
                                                   const float* __restrict__ ew,
                                                   float* __restrict__ deg, int e) {
  int i = blockIdx.x * blockDim.x + threadIdx.x;
  if (i < e) atom_add_f32(&deg[col[i]], ew[i]);
}

__global__ __launch_bounds__(256) void k_dinv(const float* __restrict__ deg,
                                              float* __restrict__ dinv, int n) {
  int i = blockIdx.x * blockDim.x + threadIdx.x;
  if (i < n) {
    float d = deg[i];
    dinv[i] = (d > 0.0f) ? rsqrtf(d) : 0.0f;
  }
}

__global__ __launch_bounds__(256) void k_norm(const int* __restrict__ row,
                                              const int* __restrict__ col,
                                              const float* __restrict__ ew,
                                              const float* __restrict__ dinv,
                                              float* __restrict__ nrm, int e) {
  int i = blockIdx.x * blockDim.x + threadIdx.x;
  if (i < e) nrm[i] = dinv[row[i]] * ew[i] * dinv[col[i]];
}

// ---------- propagation ----------
// hout[i,:] = dinv[i]^2 * hin[i,:]   (self-loop term; also fully initializes hout)
__global__ __launch_bounds__(256) void k_selfloop(const float* __restrict__ hin,
                                                  const float* __restrict__ dinv,
                                                  float* __restrict__ hout,
                                                  long long total4) {
  long long i = (long long)blockIdx.x * blockDim.x + threadIdx.x;
  if (i >= total4) return;
  int node = (int)(i >> 5);  // 32 float4 per 128-float row
  float dd = dinv[node];
  float s = dd * dd;
  float4 v = ((const float4*)hin)[i];
  float4 r;
  r.x = v.x * s; r.y = v.y * s; r.z = v.z * s; r.w = v.w * s;
  ((float4*)hout)[i] = r;
}

// one wave per edge: coalesced 512B row gather, native f32 atomic scatter
__global__ __launch_bounds__(256) void k_spmm(const int* __restrict__ row,
                                              const int* __restrict__ col,
                                              const float* __restrict__ nrm,
                                              const float* __restrict__ hin,
                                              float* __restrict__ hout, int e) {
  int gid = blockIdx.x * blockDim.x + threadIdx.x;
  int eid = gid >> 5;
  int lane = threadIdx.x & 31;
  if (eid >= e) return;
  int r = row[eid];
  int c = col[eid];
  float w = nrm[eid];
  float4 v = ((const float4*)(hin + (size_t)r * D))[lane];
  float* dst = hout + (size_t)c * D + lane * 4;
  atom_add_f32(dst + 0, w * v.x);
  atom_add_f32(dst + 1, w * v.y);
  atom_add_f32(dst + 2, w * v.z);
  atom_add_f32(dst + 3, w * v.w);
}

// ---------- final linear: out = h @ W^T + bias via V_WMMA_F32_16X16X4_F32 ----------
// block = 256 threads = 8 waves; block handles rows [16*blockIdx.x, +16), wave w
// handles output cols [16*w, +16). A-tile (16x128) staged in LDS once per block.
__global__ __launch_bounds__(256) void k_gemm(const float* __restrict__ h,
                                              const float* __restrict__ W,
                                              const float* __restrict__ bias,
                                              float* __restrict__ out) {
  __shared__ float As[16 * D];  // 8 KB
  const int m0 = blockIdx.x * 16;

  {  // cooperative A-tile stage (2048 floats = 512 float4, 256 threads)
    const float4* src = (const float4*)(h + (size_t)m0 * D);
    float4* dst = (float4*)As;
#pragma unroll
    for (int i = threadIdx.x; i < (16 * D) / 4; i += 256) dst[i] = src[i];
  }
  __syncthreads();

  const int lane = threadIdx.x & 31;
  const int wv = threadIdx.x >> 5;
  const int n0 = wv * 16;
  const int lm = lane & 15;          // M for A-frag, N for B/C/D frags
  const int kh = (lane >> 4) << 1;   // K sub-offset: 0 (lanes 0-15) / 2 (lanes 16-31)

  const float* arow = As + lm * D;                 // A[M=lm][k]
  const float* brow = W + (size_t)(n0 + lm) * D;   // B[k][N=lm] = W[n0+lm][k]

  v8f c;
  float bv = bias[n0 + lm];  // C/D col depends only on lane -> broadcast bias
#pragma unroll
  for (int j = 0; j < 8; ++j) c[j] = bv;

#pragma unroll
  for (int k0 = 0; k0 < D; k0 += 4) {
    v2f a, b;
    a[0] = arow[k0 + kh];
    a[1] = arow[k0 + kh + 1];
    b[0] = brow[k0 + kh];
    b[1] = brow[k0 + kh + 1];
    c = __builtin_amdgcn_wmma_f32_16x16x4_f32(false, a, false, b, (short)0, c,
                                              false, false);
  }

  // D layout: VGPR j -> row m0 + j + 8*(lane>=16), col n0 + lm
  float* orow = out + (size_t)(m0 + ((lane >> 4) << 3)) * D + n0 + lm;
#pragma unroll
  for (int j = 0; j < 8; ++j) orow[(size_t)j * D] = c[j];
}

// scalar tail for row counts not divisible by 16 (unused for N=100000)
__global__ __launch_bounds__(128) void k_gemm_tail(const float* __restrict__ h,
                                                   const float* __restrict__ W,
                                                   const float* __restrict__ bias,
                                                   float* __restrict__ out,
                                                   int mstart, int mend) {
  int m = mstart + blockIdx.x;
  if (m >= mend) return;
  int o = threadIdx.x;  // 128 threads = one output col each
  float acc = bias[o];
  for (int k = 0; k < D; ++k) acc += h[(size_t)m * D + k] * W[(size_t)o * D + k];
  out[(size_t)m * D + o] = acc;
}

extern "C" void kernel_launch(void* const* d_in, const int* in_sizes, int n_in,
                              void* d_out, int out_size, void* d_ws, size_t ws_size,
                              hipStream_t stream) {
  const float* x    = (const float*)d_in[0];
  const int*   ei   = (const int*)d_in[1];
  const float* ew   = (const float*)d_in[2];
  const float* Wt   = (const float*)d_in[3];  // [D_OUT, D_IN]
  const float* bias = (const float*)d_in[4];

  const int N = in_sizes[0] / D;
  const int E = in_sizes[2];
  const int* row = ei;       // sources
  const int* col = ei + E;   // targets

  // workspace carve-up (256B aligned): deg | dinv | norm | h2
  char* ws = (char*)d_ws;
  size_t off = 0;
  auto align256 = [](size_t v) { return (v + 255) & ~(size_t)255; };
  float* deg  = (float*)(ws + off); off += align256((size_t)N * 4);
  float* dinv = (float*)(ws + off); off += align256((size_t)N * 4);
  float* nrm  = (float*)(ws + off); off += align256((size_t)E * 4);
  float* h2   = (float*)(ws + off); off += align256((size_t)N * D * 4);
  (void)ws_size;

  float* h1 = (float*)d_out;  // reuse output buffer as hop-1 result

  dim3 blk(256);
  int gN  = (N + 255) / 256;
  int gE  = (E + 255) / 256;
  long long nv4 = (long long)N * (D / 4);
  int gNV = (int)((nv4 + 255) / 256);
  int gEw = (int)(((long long)E * 32 + 255) / 256);  // one wave per edge

  // --- gcn_norm ---
  k_init_deg<<<gN, blk, 0, stream>>>(deg, N);
  k_accum_deg<<<gE, blk, 0, stream>>>(col, ew, deg, E);
  k_dinv<<<gN, blk, 0, stream>>>(deg, dinv, N);
  k_norm<<<gE, blk, 0, stream>>>(row, col, ew, dinv, nrm, E);

  // --- hop 1: x -> h1 ---
  k_selfloop<<<gNV, blk, 0, stream>>>(x, dinv, h1, nv4);
  k_spmm<<<gEw, blk, 0, stream>>>(row, col, nrm, x, h1, E);

  // --- hop 2: h1 -> h2 ---
  k_selfloop<<<gNV, blk, 0, stream>>>(h1, dinv, h2, nv4);
  k_spmm<<<gEw, blk, 0, stream>>>(row, col, nrm, h1, h2, E);

  // --- final linear: out = h2 @ W^T + bias ---
  int mfull = N / 16;
  if (mfull > 0) k_gemm<<<mfull, blk, 0, stream>>>(h2, Wt, bias, (float*)d_out);
  int rem = N - mfull * 16;
  if (rem > 0)
    k_gemm_tail<<<rem, dim3(128), 0, stream>>>(h2, Wt, bias, (float*)d_out,
                                               mfull * 16, N);
}